// LBPSemanticDependency_33904471834781
// MI455X (gfx1250) — compile-verified
//
#include <hip/hip_runtime.h>

// LBP semantic-dependency marginals, MI455X (gfx1250, wave32).
// One workgroup (1024 thr = 32 waves) per (b, x) sub-problem.
// - Scores for the sub-problem staged once into LDS (196KB of the 320KB WGP
//   LDS) via GLOBAL_LOAD_ASYNC_TO_LDS_B128 (ASYNCcnt path).
// - Per-pair message state reduced to the label-difference delta
//   (3 floats/pair, VGPR-resident across all 3 iterations).
// - y-reduction via V_WMMA_F32_16X16X4_F32: A = partials in the documented
//   16x4 f32 A layout, B = all-ones (layout-independent), accumulate 2 chunks.

#define SDIM  128
#define NITER 3

typedef float v2f __attribute__((ext_vector_type(2)));
typedef float v8f __attribute__((ext_vector_type(8)));
typedef int   v4i __attribute__((vector_size(4 * sizeof(int))));
typedef __attribute__((address_space(1))) v4i* gv4i_p;   // global v4i*
typedef __attribute__((address_space(3))) v4i* lv4i_p;   // LDS v4i*

#if defined(__has_builtin)
#  if __has_builtin(__builtin_amdgcn_global_load_async_to_lds_b128)
#    define HAVE_ASYNC_LDS 1
#  endif
#  if __has_builtin(__builtin_amdgcn_s_wait_asynccnt)
#    define WAIT_ASYNC() __builtin_amdgcn_s_wait_asynccnt(0)
#  else
#    define WAIT_ASYNC() asm volatile("s_wait_asynccnt 0x0" ::: "memory")
#  endif
#endif

__device__ __forceinline__ float softplus_f(float v) {
    // log(1 + exp(v)), numerically stable
    return fmaxf(v, 0.0f) + __logf(1.0f + __expf(-fabsf(v)));
}

__global__ void __launch_bounds__(1024)
lbp_sdp_kernel(const float* __restrict__ s_edge,
               const float* __restrict__ s_sib,
               const float* __restrict__ s_cop,
               const float* __restrict__ s_grd,
               const unsigned char* __restrict__ mask,
               float* __restrict__ out)
{
    const int wg  = blockIdx.x;            // b*SDIM + x
    const int b   = wg >> 7;
    const int x   = wg & (SDIM - 1);
    const int tid = threadIdx.x;

    extern __shared__ float smem[];
    float* sco  = smem;                    // [3][SDIM][SDIM] scores s_t[y][k]
    float* part = sco + 3 * SDIM * SDIM;   // [8][SDIM] masked partial delta sums
    float* dq   = part + 8 * SDIM;         // [SDIM]   q1-q0 per k
    float* se1  = dq + SDIM;               // [SDIM]   s_edge[b,k,x]
    float* mkx  = se1 + SDIM;              // [SDIM]   mask[b,y,x]

    // ---- Stage this (b,x)'s second-order scores into LDS ----
    // s_t[b, y, x, k]: row y = 128 contiguous floats at b*S^3 + y*S^2 + x*S
    {
        const size_t probe = ((size_t)b * SDIM * SDIM + (size_t)x) * SDIM;
        for (int idx = tid; idx < 3 * SDIM * (SDIM / 4); idx += 1024) {
            const int f4i = idx & (SDIM / 4 - 1);   // 0..31
            const int row = idx >> 5;               // 0..383
            const int ty  = row >> 7;               // tensor 0..2
            const int y   = row & (SDIM - 1);
            const float* src = (ty == 0) ? s_sib : ((ty == 1) ? s_cop : s_grd);
            const float* g = src + probe + (size_t)y * SDIM * SDIM + (size_t)f4i * 4;
            float*       l = sco + (ty * SDIM + y) * SDIM + f4i * 4;
#ifdef HAVE_ASYNC_LDS
            __builtin_amdgcn_global_load_async_to_lds_b128((gv4i_p)g, (lv4i_p)l,
                                                           0, 0);
#else
            *(float4*)l = *(const float4*)g;
#endif
        }
#ifdef HAVE_ASYNC_LDS
        WAIT_ASYNC();
#endif
    }
    if (tid < SDIM) {
        const size_t off = (((size_t)b * SDIM) + (size_t)tid) * SDIM + x; // (b,tid,x)
        const float e = s_edge[off];
        se1[tid] = e;
        mkx[tid] = mask[off] ? 1.0f : 0.0f;
        dq[tid]  = e;                       // initial q = se  ->  q1-q0 = s_edge
    }
    __syncthreads();

    // ---- Per-thread pair set: fixed k, 16 consecutive y ----
    const int k     = tid & (SDIM - 1);
    const int y8    = tid >> 7;            // 0..7
    const int ybase = y8 * 16;

    const float* sc_s = sco + 0 * SDIM * SDIM + k;
    const float* sc_c = sco + 1 * SDIM * SDIM + k;
    const float* sc_g = sco + 2 * SDIM * SDIM + k;
    const float  mk_k = mkx[k];

    float dsib[16], dcop[16], dgrd[16];
    #pragma unroll
    for (int j = 0; j < 16; ++j) { dsib[j] = 0.f; dcop[j] = 0.f; dgrd[j] = 0.f; }

    for (int it = 0; it < NITER; ++it) {
        // ---- message updates (delta form) + masked partial sum ----
        float acc = 0.0f;
        #pragma unroll
        for (int j = 0; j < 16; ++j) {
            const int   y   = ybase + j;
            const float dqy = dq[y];
            float u, d, tot;

            u = dqy - dsib[j];
            d = softplus_f(u + sc_s[y * SDIM]) - softplus_f(u);
            dsib[j] = d; tot = d;

            u = dqy - dcop[j];
            d = softplus_f(u + sc_c[y * SDIM]) - softplus_f(u);
            dcop[j] = d; tot += d;

            u = dqy - dgrd[j];
            d = softplus_f(u + sc_g[y * SDIM]) - softplus_f(u);
            dgrd[j] = d; tot += d;

            const float mval = ((y != x) && (y != k)) ? mkx[y] : 0.0f;
            acc += mval * tot;
        }
        part[y8 * SDIM + k] = acc * mk_k;
        __syncthreads();

        // ---- dq[k] = se1[k] + sum_{y8} part[y8][k] via WMMA (waves 0..7) ----
        // Recompute all indices locally from threadIdx so nothing stays live
        // (and gets spilled) across the big message loop above.
        if (threadIdx.x < 256) {
            const int lane  = (int)(threadIdx.x & 31u);   // wave32
            const int kbase = (int)(threadIdx.x >> 5) * 16;
            const int m_    = lane & 15;          // A-layout: M = lane&15
            const int kk    = (lane >> 4) * 2;    // A VGPR0: K=0/2, VGPR1: K=1/3
            v8f c = {};
            v2f bb; bb[0] = 1.0f; bb[1] = 1.0f;   // B = ones (layout-free)
            v2f a;
            a[0] = part[(kk + 0) * SDIM + kbase + m_];
            a[1] = part[(kk + 1) * SDIM + kbase + m_];
            c = __builtin_amdgcn_wmma_f32_16x16x4_f32(false, a, false, bb,
                                                      (short)0, c, false, false);
            a[0] = part[(kk + 4) * SDIM + kbase + m_];
            a[1] = part[(kk + 5) * SDIM + kbase + m_];
            c = __builtin_amdgcn_wmma_f32_16x16x4_f32(false, a, false, bb,
                                                      (short)0, c, false, false);
            // C layout: lane with N==0 holds M=v (lane 0) / M=v+8 (lane 16)
            if ((lane & 15) == 0) {
                const int mo = (lane >> 4) * 8;
                #pragma unroll
                for (int v = 0; v < 8; ++v) {
                    const int kidx = kbase + mo + v;
                    dq[kidx] = se1[kidx] + c[v];
                }
            }
        }
        __syncthreads();
    }

    // ---- out[b, k, x] = sigmoid(dq[k]) ----
    if (tid < SDIM) {
        const size_t off = (((size_t)b * SDIM) + (size_t)tid) * SDIM + x;
        out[off] = __builtin_amdgcn_rcpf(1.0f + __expf(-dq[tid]));
    }
}

extern "C" void kernel_launch(void* const* d_in, const int* in_sizes, int n_in,
                              void* d_out, int out_size, void* d_ws, size_t ws_size,
                              hipStream_t stream) {
    (void)n_in; (void)d_ws; (void)ws_size; (void)out_size;
    const float* s_edge = (const float*)d_in[0];
    const float* s_sib  = (const float*)d_in[1];
    const float* s_cop  = (const float*)d_in[2];
    const float* s_grd  = (const float*)d_in[3];
    const unsigned char* mask = (const unsigned char*)d_in[4];
    float* out = (float*)d_out;

    const int nblk = in_sizes[0] / SDIM;   // B * S workgroups, one per (b, x)
    const size_t shmem =
        (size_t)(3 * SDIM * SDIM + 8 * SDIM + 3 * SDIM) * sizeof(float); // ~202 KB

    (void)hipFuncSetAttribute((const void*)lbp_sdp_kernel,
                              hipFuncAttributeMaxDynamicSharedMemorySize,
                              (int)shmem);
    lbp_sdp_kernel<<<dim3(nblk), dim3(1024), shmem, stream>>>(
        s_edge, s_sib, s_cop, s_grd, mask, out);
}